// AIRSGNN_86217173500361
// MI455X (gfx1250) — compile-verified
//
#include <hip/hip_runtime.h>
#include <math.h>

#define HIDN 256
#define NHEADS 4
#define CH 64
#define FDIM 64
#define NLAYERS 4
#define NGRAPHS 16

typedef __attribute__((ext_vector_type(2))) float v2f;
typedef __attribute__((ext_vector_type(8))) float v8f;

// ---------------- monotone float<->uint encoding for atomicMax on floats ----
__device__ __forceinline__ unsigned encf(float f) {
    unsigned u = __float_as_uint(f);
    return (u & 0x80000000u) ? ~u : (u | 0x80000000u);
}
__device__ __forceinline__ float decf(unsigned u) {
    u = (u & 0x80000000u) ? (u & 0x7FFFFFFFu) : ~u;
    return __uint_as_float(u);
}
#define ENC_NEG_INF 0x007FFFFFu  // encf(-inf)

// ---------------- feats = [x | region_emb[region_ids] | posenc] -------------
__global__ __launch_bounds__(256) void k_feats(const float* __restrict__ x,
                                               const float* __restrict__ remb,
                                               const int* __restrict__ rid,
                                               float* __restrict__ feats, int nN) {
    int idx = blockIdx.x * blockDim.x + threadIdx.x;
    if (idx >= nN * 192) return;
    int n = idx / 192, c = idx - n * 192;
    float v;
    if (c < 64) {
        v = x[(size_t)n * 64 + c];
    } else if (c < 128) {
        v = remb[(size_t)rid[n] * 64 + (c - 64)];
    } else {
        int i = c - 128;
        float rate = powf(10000.0f, -2.0f * floorf(i * 0.5f) / 64.0f);
        float ang = (float)n * rate;
        v = ((i & 1) == 0) ? sinf(ang) : cosf(ang);
    }
    feats[idx] = v;
}

// ---------------- fp32 WMMA GEMM: C = A[M,K] @ W[K,N] (+bias) ---------------
// One wave32 per block. Wave computes a (16*MT)x64 tile: MT m-subtiles reuse the
// same 4 B-fragments per k-step -> 16 WMMAs per 12 VMEM issues at MT=4.
// Requires: rows covered are in-range (host splits M into MT=4 body + MT=1 tail),
// N % 64 == 0, K % 4 == 0. No divergence: EXEC all-ones around every WMMA.
template <int MT>
__global__ __launch_bounds__(32) void k_gemm_wmma(const float* __restrict__ A,
                                                  const float* __restrict__ W,
                                                  const float* __restrict__ bias,
                                                  float* __restrict__ Cout,
                                                  int mbase, int K, int N) {
    const int lane = threadIdx.x & 31;
    const int m0 = mbase + blockIdx.x * (16 * MT);
    const int n0 = blockIdx.y * 64;
    const bool hi = lane >= 16;
    const int l16 = lane & 15;
    const int koff = hi ? 2 : 0;

    v8f acc[MT][4];
#pragma unroll
    for (int mt = 0; mt < MT; ++mt)
#pragma unroll
        for (int t = 0; t < 4; ++t) acc[mt][t] = (v8f){};

    const float* arow[MT];
#pragma unroll
    for (int mt = 0; mt < MT; ++mt)
        arow[mt] = A + (size_t)(m0 + mt * 16 + l16) * K + koff;

    for (int k = 0; k < K; k += 4) {
        // A fragments: v0 = A[m, k+koff], v1 = A[m, k+koff+1]
        v2f a[MT];
#pragma unroll
        for (int mt = 0; mt < MT; ++mt) {
            float2 af = *reinterpret_cast<const float2*>(arow[mt] + k);
            a[mt].x = af.x; a[mt].y = af.y;
        }
        // B fragments: v0 = W[k+koff, col], v1 = W[k+koff+1, col]
        const float* w0 = W + (size_t)(k + koff) * N + n0 + l16;
        const float* w1 = w0 + N;
        v2f b[4];
#pragma unroll
        for (int t = 0; t < 4; ++t) { b[t].x = w0[t * 16]; b[t].y = w1[t * 16]; }
#pragma unroll
        for (int t = 0; t < 4; ++t)
#pragma unroll
            for (int mt = 0; mt < MT; ++mt)
                acc[mt][t] = __builtin_amdgcn_wmma_f32_16x16x4_f32(
                    false, a[mt], false, b[t], (short)0, acc[mt][t], false, false);
    }
    // C/D layout: vgpr i -> row sub+i (lanes 0-15) or sub+8+i (lanes 16-31)
#pragma unroll
    for (int mt = 0; mt < MT; ++mt) {
        const int rbase = m0 + mt * 16 + (hi ? 8 : 0);
#pragma unroll
        for (int t = 0; t < 4; ++t) {
            int col = n0 + t * 16 + l16;
            float bv = bias ? bias[col] : 0.0f;
#pragma unroll
            for (int i = 0; i < 8; ++i)
                Cout[(size_t)(rbase + i) * N + col] = acc[mt][t][i] + bv;
        }
    }
}

// ---------------- per-(node,head) attention scores --------------------------
__global__ __launch_bounds__(256) void k_scores(const float* __restrict__ hproj,
                                                const float* __restrict__ asrc,
                                                const float* __restrict__ adst,
                                                float* __restrict__ ssrc,
                                                float* __restrict__ sdst, int nN) {
    int idx = blockIdx.x * blockDim.x + threadIdx.x;
    if (idx >= nN * NHEADS) return;
    int n = idx >> 2, hd = idx & 3;
    const float* hp = hproj + (size_t)n * HIDN + hd * CH;
    const float* as = asrc + hd * CH;
    const float* ad = adst + hd * CH;
    float ss = 0.f, sd = 0.f;
#pragma unroll 4
    for (int c = 0; c < CH; ++c) { float v = hp[c]; ss += v * as[c]; sd += v * ad[c]; }
    ssrc[idx] = ss; sdst[idx] = sd;
}

__global__ __launch_bounds__(256) void k_init_m(unsigned* __restrict__ m, int n) {
    int i = blockIdx.x * blockDim.x + threadIdx.x;
    if (i < n) m[i] = ENC_NEG_INF;
}

// ---------------- edge pass A: e = leaky_relu(s_src+s_dst); segment max -----
__global__ __launch_bounds__(256) void k_edge_a(const int* __restrict__ esrc,
                                                const int* __restrict__ edst,
                                                const float* __restrict__ ssrc,
                                                const float* __restrict__ sdst,
                                                float* __restrict__ ebuf,
                                                unsigned* __restrict__ menc,
                                                int nE, int nN) {
    int idx = blockIdx.x * blockDim.x + threadIdx.x;
    if (idx >= (nE + nN) * NHEADS) return;
    int e = idx >> 2, hd = idx & 3;
    int s = (e < nE) ? esrc[e] : (e - nE);
    int d = (e < nE) ? edst[e] : (e - nE);
    float v = ssrc[s * NHEADS + hd] + sdst[d * NHEADS + hd];
    v = (v > 0.f) ? v : 0.2f * v;   // LeakyReLU(0.2)
    ebuf[idx] = v;
    atomicMax(&menc[d * NHEADS + hd], encf(v));
}

// ---------------- edge pass B: ex = exp(e - max); segment sum ---------------
__global__ __launch_bounds__(256) void k_edge_b(const int* __restrict__ edst,
                                                float* __restrict__ ebuf,
                                                const unsigned* __restrict__ menc,
                                                float* __restrict__ denom,
                                                int nE, int nN) {
    int idx = blockIdx.x * blockDim.x + threadIdx.x;
    if (idx >= (nE + nN) * NHEADS) return;
    int e = idx >> 2, hd = idx & 3;
    int d = (e < nE) ? edst[e] : (e - nE);
    float m = decf(menc[d * NHEADS + hd]);
    float ex = expf(ebuf[idx] - m);
    ebuf[idx] = ex;
    atomicAdd(&denom[d * NHEADS + hd], ex);
}

// ---------------- edge pass C: agg[dst] += alpha * hproj[src] ---------------
__global__ __launch_bounds__(256) void k_edge_c(const int* __restrict__ esrc,
                                                const int* __restrict__ edst,
                                                const float* __restrict__ ebuf,
                                                const float* __restrict__ denom,
                                                const float* __restrict__ hproj,
                                                float* __restrict__ agg,
                                                int nE, int nN) {
    int idx = blockIdx.x * blockDim.x + threadIdx.x;
    if (idx >= (nE + nN) * NHEADS) return;
    int e = idx >> 2, hd = idx & 3;
    int s = (e < nE) ? esrc[e] : (e - nE);
    int d = (e < nE) ? edst[e] : (e - nE);
    float alpha = ebuf[idx] / denom[d * NHEADS + hd];
    const float4* hs = reinterpret_cast<const float4*>(hproj + (size_t)s * HIDN + hd * CH);
    float* op = agg + (size_t)d * HIDN + hd * CH;
#pragma unroll
    for (int q = 0; q < CH / 4; ++q) {
        float4 hv = hs[q];
        atomicAdd(&op[q * 4 + 0], alpha * hv.x);
        atomicAdd(&op[q * 4 + 1], alpha * hv.y);
        atomicAdd(&op[q * 4 + 2], alpha * hv.z);
        atomicAdd(&op[q * 4 + 3], alpha * hv.w);
    }
}

// ---------------- bias + relu + layernorm (one wave32 per node) -------------
__global__ __launch_bounds__(256) void k_bias_relu_ln(const float* __restrict__ agg,
                                                      const float* __restrict__ gbias,
                                                      const float* __restrict__ g,
                                                      const float* __restrict__ bta,
                                                      float* __restrict__ out, int nN) {
    int node = blockIdx.x * 8 + (threadIdx.x >> 5);
    int lane = threadIdx.x & 31;
    if (node >= nN) return;
    const float* row = agg + (size_t)node * HIDN;
    float v[8];
    float s = 0.f;
#pragma unroll
    for (int j = 0; j < 8; ++j) {
        int c = lane * 8 + j;
        float t = fmaxf(row[c] + gbias[c], 0.f);
        v[j] = t; s += t;
    }
#pragma unroll
    for (int o = 16; o > 0; o >>= 1) s += __shfl_xor(s, o, 32);
    float mu = s * (1.0f / HIDN);
    float q = 0.f;
#pragma unroll
    for (int j = 0; j < 8; ++j) { float d = v[j] - mu; q += d * d; }
#pragma unroll
    for (int o = 16; o > 0; o >>= 1) q += __shfl_xor(q, o, 32);
    float rstd = rsqrtf(q * (1.0f / HIDN) + 1e-5f);
    float* orow = out + (size_t)node * HIDN;
#pragma unroll
    for (int j = 0; j < 8; ++j) {
        int c = lane * 8 + j;
        orow[c] = (v[j] - mu) * rstd * g[c] + bta[c];
    }
}

// ---------------- mean pooling: LDS partials, then global atomics -----------
#define POOL_CHUNK 512
__global__ __launch_bounds__(256) void k_pool(const float* __restrict__ h,
                                              const int* __restrict__ batch,
                                              float* __restrict__ sums,
                                              float* __restrict__ cnts, int nN) {
    __shared__ float acc[NGRAPHS * HIDN];
    __shared__ float cacc[NGRAPHS];
    int tid = threadIdx.x;
    for (int i = tid; i < NGRAPHS * HIDN; i += 256) acc[i] = 0.f;
    if (tid < NGRAPHS) cacc[tid] = 0.f;
    __syncthreads();
    int start = blockIdx.x * POOL_CHUNK;
    int end = min(start + POOL_CHUNK, nN);
    for (int n = start; n < end; ++n) {
        int b = batch[n];
        acc[b * HIDN + tid] += h[(size_t)n * HIDN + tid];
        if (tid == 0) cacc[b] += 1.f;
    }
    __syncthreads();
    for (int i = tid; i < NGRAPHS * HIDN; i += 256) atomicAdd(&sums[i], acc[i]);
    if (tid < NGRAPHS) atomicAdd(&cnts[tid], cacc[tid]);
}

// ---------------- readout MLP (tiny: 16x256 @ 256x256) ----------------------
__global__ __launch_bounds__(HIDN) void k_readout1(const float* __restrict__ sums,
                                                   const float* __restrict__ cnts,
                                                   const float* __restrict__ W1,
                                                   const float* __restrict__ b1,
                                                   float* __restrict__ t1) {
    int b = blockIdx.x, c = threadIdx.x;
    float cnt = fmaxf(cnts[b], 1.0f);
    float inv = 1.0f / cnt;
    float acc = b1[c];
    for (int k = 0; k < HIDN; ++k) acc += (sums[b * HIDN + k] * inv) * W1[k * HIDN + c];
    // exact GELU: 0.5*x*(1+erf(x/sqrt(2)))
    t1[b * HIDN + c] = 0.5f * acc * (1.0f + erff(acc * 0.70710678118654752f));
}

__global__ __launch_bounds__(HIDN) void k_readout2(const float* __restrict__ t1,
                                                   const float* __restrict__ W2,
                                                   const float* __restrict__ b2,
                                                   float* __restrict__ out) {
    int b = blockIdx.x, c = threadIdx.x;
    float acc = b2[c];
    for (int k = 0; k < HIDN; ++k) acc += t1[b * HIDN + k] * W2[k * HIDN + c];
    out[b * HIDN + c] = acc;
}

// ---------------------------------------------------------------------------
extern "C" void kernel_launch(void* const* d_in, const int* in_sizes, int n_in,
                              void* d_out, int out_size, void* d_ws, size_t ws_size,
                              hipStream_t stream) {
    const float* x          = (const float*)d_in[0];
    const float* remb       = (const float*)d_in[1];
    const float* in_proj_w  = (const float*)d_in[2];
    const float* in_proj_b  = (const float*)d_in[3];
    const float* gat_w      = (const float*)d_in[4];
    const float* att_src    = (const float*)d_in[5];
    const float* att_dst    = (const float*)d_in[6];
    const float* gat_b      = (const float*)d_in[7];
    const float* ln_g       = (const float*)d_in[8];
    const float* ln_b       = (const float*)d_in[9];
    const float* ro_w1      = (const float*)d_in[10];
    const float* ro_b1      = (const float*)d_in[11];
    const float* ro_w2      = (const float*)d_in[12];
    const float* ro_b2      = (const float*)d_in[13];
    const int*   edge_index = (const int*)d_in[14];
    const int*   batch      = (const int*)d_in[15];
    const int*   rid        = (const int*)d_in[16];

    const int nN = in_sizes[15];
    const int nE = in_sizes[14] / 2;
    const int* esrc = edge_index;
    const int* edst = edge_index + nE;

    char* ws = (char*)d_ws;
    size_t off = 0;
    auto alloc = [&](size_t bytes) -> char* {
        char* p = ws + off;
        off += (bytes + 255) & ~(size_t)255;
        return p;
    };
    float*    feats = (float*)alloc((size_t)nN * 192 * 4);
    float*    h     = (float*)alloc((size_t)nN * HIDN * 4);
    float*    hproj = (float*)alloc((size_t)nN * HIDN * 4);
    float*    agg   = (float*)alloc((size_t)nN * HIDN * 4);
    float*    ssrc  = (float*)alloc((size_t)nN * NHEADS * 4);
    float*    sdst  = (float*)alloc((size_t)nN * NHEADS * 4);
    unsigned* menc  = (unsigned*)alloc((size_t)nN * NHEADS * 4);
    float*    denom = (float*)alloc((size_t)nN * NHEADS * 4);
    float*    ebuf  = (float*)alloc((size_t)(nE + nN) * NHEADS * 4);
    float*    sums  = (float*)alloc(NGRAPHS * HIDN * 4);
    float*    cnts  = (float*)alloc(NGRAPHS * 4);
    float*    t1    = (float*)alloc(NGRAPHS * HIDN * 4);
    (void)ws_size; (void)n_in; (void)out_size;

    // WMMA GEMM launcher: MT=4 body (64 rows/wave) + MT=1 tail (16 rows/wave)
    auto launch_gemm = [&](const float* A, const float* W, const float* bias,
                           float* C, int M, int K) {
        int body = M / 64;               // blocks of 64 rows
        int done = body * 64;
        if (body > 0) {
            dim3 g(body, HIDN / 64);
            k_gemm_wmma<4><<<g, 32, 0, stream>>>(A, W, bias, C, 0, K, HIDN);
        }
        int tail = (M - done) / 16;      // remaining 16-row blocks (M % 16 == 0)
        if (tail > 0) {
            dim3 g(tail, HIDN / 64);
            k_gemm_wmma<1><<<g, 32, 0, stream>>>(A, W, bias, C, done, K, HIDN);
        }
    };

    // 1) build input features and project to HID
    {
        int total = nN * 192;
        k_feats<<<(total + 255) / 256, 256, 0, stream>>>(x, remb, rid, feats, nN);
        launch_gemm(feats, in_proj_w, in_proj_b, h, nN, 3 * FDIM);
    }

    const int ETOT = (nE + nN) * NHEADS;
    const int nh = nN * NHEADS;
    for (int l = 0; l < NLAYERS; ++l) {
        const float* Wl = gat_w + (size_t)l * HIDN * HIDN;
        launch_gemm(h, Wl, nullptr, hproj, nN, HIDN);
        k_scores<<<(nh + 255) / 256, 256, 0, stream>>>(
            hproj, att_src + l * NHEADS * CH, att_dst + l * NHEADS * CH, ssrc, sdst, nN);
        k_init_m<<<(nh + 255) / 256, 256, 0, stream>>>(menc, nh);
        hipMemsetAsync(denom, 0, (size_t)nh * 4, stream);
        hipMemsetAsync(agg, 0, (size_t)nN * HIDN * 4, stream);
        k_edge_a<<<(ETOT + 255) / 256, 256, 0, stream>>>(esrc, edst, ssrc, sdst, ebuf, menc, nE, nN);
        k_edge_b<<<(ETOT + 255) / 256, 256, 0, stream>>>(edst, ebuf, menc, denom, nE, nN);
        k_edge_c<<<(ETOT + 255) / 256, 256, 0, stream>>>(esrc, edst, ebuf, denom, hproj, agg, nE, nN);
        k_bias_relu_ln<<<(nN + 7) / 8, 256, 0, stream>>>(
            agg, gat_b + l * HIDN, ln_g + l * HIDN, ln_b + l * HIDN, h, nN);
    }

    // 2) mean pool + readout MLP
    hipMemsetAsync(sums, 0, NGRAPHS * HIDN * 4, stream);
    hipMemsetAsync(cnts, 0, NGRAPHS * 4, stream);
    k_pool<<<(nN + POOL_CHUNK - 1) / POOL_CHUNK, 256, 0, stream>>>(h, batch, sums, cnts, nN);
    k_readout1<<<NGRAPHS, HIDN, 0, stream>>>(sums, cnts, ro_w1, ro_b1, t1);
    k_readout2<<<NGRAPHS, HIDN, 0, stream>>>(t1, ro_w2, ro_b2, (float*)d_out);
}